// Reasoning_layer_86096914416018
// MI455X (gfx1250) — compile-verified
//
#include <hip/hip_runtime.h>
#include <hip/hip_bf16.h>
#include <math.h>

// ---------------------------------------------------------------------------
// Sizes (compile-time constants from the reference)
// ---------------------------------------------------------------------------
#define BB   2
#define NN_  32
#define DD   768
#define HH   4
#define HDIM 192
#define ROWS (BB * NN_ * NN_)        // 2048 flattened (b,i,j) rows
#define SEL  (ROWS * DD)             // 1,572,864 elements per full buffer
#define SPH  (ROWS * HDIM)           // per-head buffer elements (SEL/4)

// GEMM tiling
#define BM    128
#define BN    64
#define BKC   32
#define LDA_S 36   // LDS row stride (floats) for A tile: 16B-aligned rows, bank-spread
#define LDB_S 68   // LDS row stride (floats) for B tile: 16B-aligned rows, bank-spread

typedef __attribute__((ext_vector_type(2))) float v2f;
typedef __attribute__((ext_vector_type(8))) float v8f;

#if defined(__gfx1250__) && __has_builtin(__builtin_amdgcn_global_load_async_to_lds_b128) && __has_builtin(__builtin_amdgcn_s_wait_asynccnt)
#define USE_ASYNC_LDS 1
typedef int v4i_vs __attribute__((vector_size(16)));
typedef __attribute__((address_space(1))) v4i_vs* g_v4i_ptr;  // prints as "__device__"
typedef __attribute__((address_space(3))) v4i_vs* l_v4i_ptr;  // prints as "__shared__"
#endif

__device__ __forceinline__ void stage16B(const float* gp, float* lp) {
#ifdef USE_ASYNC_LDS
    __builtin_amdgcn_global_load_async_to_lds_b128((g_v4i_ptr)gp, (l_v4i_ptr)lp, 0, 0);
#else
    *(float4*)lp = *(const float4*)gp;
#endif
}

__device__ __forceinline__ void stage_wait() {
#ifdef USE_ASYNC_LDS
    __builtin_amdgcn_s_wait_asynccnt(0);
#endif
}

// ---------------------------------------------------------------------------
// LDS-blocked f32 WMMA GEMM:  C[M,N] = A[M,K] * B[K,N] (+bias) (*rowmask)
// A has leading dim lda and column offset aoff (slices one head out of K/V).
// Block = 256 threads = 8 waves. Tile BM=128 x BN=64, K stepped by BKC=32.
// Wave w computes rows [w*16, w*16+16) x all 64 cols => 4 accumulators,
// so each A fragment feeds 4 WMMAs and each staged B tile feeds 8 waves.
//
// Fragment layouts (CDNA5 ISA 7.12.2, 32-bit data, wave32):
//   A 16x4 : lanes 0-15 -> M=lane,   VGPR0=K0, VGPR1=K1
//            lanes16-31 -> M=lane-16,VGPR0=K2, VGPR1=K3
//   B 4x16 : VGPR r holds rows K=r (lanes0-15) and K=r+2 (lanes16-31)
//   C 16x16: VGPR r holds M=r (lanes0-15) and M=r+8 (lanes16-31), N=lane&15
// ---------------------------------------------------------------------------
__global__ __launch_bounds__(256)
void gemm_f32_wmma_kernel(const float* __restrict__ A, int lda, int aoff,
                          const float* __restrict__ B, int ldb,
                          const float* __restrict__ bias,      // [N] or null
                          const float* __restrict__ rowmask,   // [M] or null
                          float* __restrict__ C, int ldc,
                          int M, int N, int K)
{
    __shared__ float As[BM * LDA_S];   // 18.0 KB
    __shared__ float Bs[BKC * LDB_S];  //  8.5 KB

    const int t    = threadIdx.x;
    const int wave = t >> 5;
    const int lane = t & 31;
    const int row0 = blockIdx.x * BM;
    const int col0 = blockIdx.y * BN;

    const int m  = lane & 15;          // M index within fragment
    const int kh = (lane >> 4) << 1;   // K offset: 0 (lanes 0-15) / 2 (lanes 16-31)
    const int nn = lane & 15;          // N index within fragment

    // staging layout: A tile 128x32 -> thread t does row t/2, 16-float half-row
    const int a_tr = t >> 1;
    const int a_tc = (t & 1) * 16;
    const float* Abase = A + (size_t)row0 * lda + aoff;

    v8f acc[4] = {};

    for (int k0 = 0; k0 < K; k0 += BKC) {
        __syncthreads();   // previous chunk fully consumed before restaging

        // ---- stage A tile (BM x BKC): 4 x 16B per thread
        {
            const float* gp = Abase + (size_t)a_tr * lda + k0 + a_tc;
            float* lp = &As[a_tr * LDA_S + a_tc];
#pragma unroll
            for (int s = 0; s < 4; ++s)
                stage16B(gp + 4 * s, lp + 4 * s);
        }
        // ---- stage B tile (BKC x BN): 2 x 16B per thread
        {
#pragma unroll
            for (int s = 0; s < 2; ++s) {
                const int q    = t + 256 * s;     // float4 linear id, 512 total
                const int brow = q >> 4;          // 16 float4 per 64-col row
                const int bcol = (q & 15) * 4;
                const float* gp = B + (size_t)(k0 + brow) * ldb + col0 + bcol;
                stage16B(gp, &Bs[brow * LDB_S + bcol]);
            }
        }
        stage_wait();
        __syncthreads();

        // ---- compute: 8 k-steps x 4 N-tiles = 32 WMMA per chunk
#pragma unroll
        for (int ks = 0; ks < BKC; ks += 4) {
            v2f a;
            const float* ap = &As[(wave * 16 + m) * LDA_S + ks + kh];
            a.x = ap[0];
            a.y = ap[1];
#pragma unroll
            for (int tn = 0; tn < 4; ++tn) {
                v2f b;
                const float* bp = &Bs[(ks + kh) * LDB_S + tn * 16 + nn];
                b.x = bp[0];
                b.y = bp[LDB_S];
                acc[tn] = __builtin_amdgcn_wmma_f32_16x16x4_f32(
                              false, a, false, b, (short)0, acc[tn], false, false);
            }
        }
    }

    // ---- epilogue
    const int rbase = (lane >> 4) * 8;
#pragma unroll
    for (int tn = 0; tn < 4; ++tn) {
        const int col = col0 + tn * 16 + nn;
        const float bv = bias ? bias[col] : 0.0f;
#pragma unroll
        for (int r = 0; r < 8; ++r) {
            const int row = row0 + wave * 16 + rbase + r;
            float v = acc[tn][r] + bv;
            if (rowmask) v *= (rowmask[row] > 0.5f) ? 1.0f : 0.0f;
            C[(size_t)row * ldc + col] = v;
        }
    }
}

// ---------------------------------------------------------------------------
// Attention over 33 keys per (query,head). One wave per query-head.
// Keys k>=1 are KA[rowA(k)] + KB[rowB(k)] (biases already folded into KA/VA).
// Lane l owns feature dims f = l, l+32, ..., l+160 (6 values of HD=192).
// Score for key k lives in lane k after wave reduction.
// ---------------------------------------------------------------------------
__device__ __forceinline__ float wred_sum(float v) {
#pragma unroll
    for (int o = 16; o > 0; o >>= 1) v += __shfl_xor(v, o, 32);
    return v;
}
__device__ __forceinline__ float wred_max(float v) {
#pragma unroll
    for (int o = 16; o > 0; o >>= 1) v = fmaxf(v, __shfl_xor(v, o, 32));
    return v;
}

__global__ __launch_bounds__(256)
void attn_kernel(const float* __restrict__ Q,
                 const float* __restrict__ Kb,
                 const float* __restrict__ Vb,
                 const float* __restrict__ KA, const float* __restrict__ KBp,
                 const float* __restrict__ VA, const float* __restrict__ VBp,
                 const float* __restrict__ mask,
                 float* __restrict__ ctx)
{
    const int lane = threadIdx.x & 31;
    const int wv   = blockIdx.x * (blockDim.x >> 5) + (threadIdx.x >> 5);
    const int head = wv & 3;
    const int qi   = wv >> 2;                 // flattened (b,i,j)
    const int b    = qi >> 10;
    const int ij   = qi & 1023;
    const int i    = ij >> 5;
    const int j    = ij & 31;
    const int b0   = b << 10;

    // index patterns per head: rowA = baseA + k*sA ; rowB = baseB + k*sB
    int baseA, sA, baseB, sB;
    switch (head) {
        case 0: baseA = b0 + i * 32; sA = 1;  baseB = b0 + j;      sB = 32; break; // ik,kj
        case 1: baseA = b0 + i * 32; sA = 1;  baseB = b0 + j * 32; sB = 1;  break; // ik,jk
        case 2: baseA = b0 + j * 32; sA = 1;  baseB = b0 + i;      sB = 32; break; // jk,ki
        default:baseA = b0 + i;      sA = 32; baseB = b0 + j;      sB = 32; break; // ki,kj
    }

    const float scale = 0.07216878364870322f;  // 1/sqrt(192)
    const float* kap = KA  + (size_t)head * SPH;
    const float* kbp = KBp + (size_t)head * SPH;
    const float* vap = VA  + (size_t)head * SPH;
    const float* vbp = VBp + (size_t)head * SPH;

    float q[6];
    const float* qp = Q + (size_t)qi * DD + head * HDIM;
#pragma unroll
    for (int u = 0; u < 6; ++u) q[u] = qp[lane + 32 * u];

    // self key (k = 0 slot, no mask add)
    const float* ksp = Kb + (size_t)qi * DD + head * HDIM;
    float part = 0.0f;
#pragma unroll
    for (int u = 0; u < 6; ++u) part += q[u] * ksp[lane + 32 * u];
    const float s0 = wred_sum(part) * scale;

    // 32 pair keys; after reduction lane k keeps score k
    float my = -3.0e38f;
    for (int k = 0; k < 32; ++k) {
        const int rA = baseA + k * sA;
        const int rB = baseB + k * sB;
        const float* pa = kap + (size_t)rA * HDIM;
        const float* pb = kbp + (size_t)rB * HDIM;
        float p = 0.0f;
#pragma unroll
        for (int u = 0; u < 6; ++u)
            p += q[u] * (pa[lane + 32 * u] + pb[lane + 32 * u]);
        float s = wred_sum(p) * scale;
        s += (1.0f - mask[rA] * mask[rB]) * -10000.0f;
        if (lane == k) my = s;
    }

    // softmax over {s0, my[0..31]}
    const float mx  = fmaxf(s0, wred_max(my));
    const float e   = __expf(my - mx);
    const float e0  = __expf(s0 - mx);
    const float den = e0 + wred_sum(e);
    const float p_l = e / den;     // prob of key (lane+1)
    const float p0  = e0 / den;    // prob of self key

    // context accumulation
    float acc[6];
    const float* vsp = Vb + (size_t)qi * DD + head * HDIM;
#pragma unroll
    for (int u = 0; u < 6; ++u) acc[u] = p0 * vsp[lane + 32 * u];

    for (int k = 0; k < 32; ++k) {
        const float pk = __shfl(p_l, k, 32);
        const int rA = baseA + k * sA;
        const int rB = baseB + k * sB;
        const float* va = vap + (size_t)rA * HDIM;
        const float* vb = vbp + (size_t)rB * HDIM;
#pragma unroll
        for (int u = 0; u < 6; ++u)
            acc[u] += pk * (va[lane + 32 * u] + vb[lane + 32 * u]);
    }

    float* op = ctx + (size_t)qi * DD + head * HDIM;
#pragma unroll
    for (int u = 0; u < 6; ++u) op[lane + 32 * u] = acc[u];
}

// ---------------------------------------------------------------------------
// Highway gate: x2 = gelu(hraw) * sigmoid(traw) + x * (1 - sigmoid(traw))
// ---------------------------------------------------------------------------
__global__ __launch_bounds__(256)
void highway_kernel(const float* __restrict__ hraw,
                    const float* __restrict__ traw,
                    const float* __restrict__ x,
                    float* __restrict__ x2, int n)
{
    const int id = blockIdx.x * blockDim.x + threadIdx.x;
    if (id >= n) return;
    const float hv = hraw[id];
    const float g  = 0.5f * hv * (1.0f + erff(hv * 0.7071067811865475f));
    const float t  = 1.0f / (1.0f + __expf(-traw[id]));
    x2[id] = g * t + x[id] * (1.0f - t);
}

// ---------------------------------------------------------------------------
// Residual + LayerNorm + node mask; writes out and the identical hidden copy.
// One 256-thread block per row of 768.
// ---------------------------------------------------------------------------
__global__ __launch_bounds__(256)
void ln_kernel(const float* __restrict__ y, const float* __restrict__ h,
               const float* __restrict__ mask,
               const float* __restrict__ g, const float* __restrict__ bta,
               float* __restrict__ out, float* __restrict__ hidden)
{
    const int r = blockIdx.x;
    const int t = threadIdx.x;
    __shared__ float red[256];
    __shared__ float s_mu, s_inv;

    float z[3];
    float s = 0.0f;
#pragma unroll
    for (int u = 0; u < 3; ++u) {
        const int c = t + 256 * u;
        z[u] = y[(size_t)r * DD + c] + h[(size_t)r * DD + c];
        s += z[u];
    }
    red[t] = s; __syncthreads();
    for (int o = 128; o > 0; o >>= 1) { if (t < o) red[t] += red[t + o]; __syncthreads(); }
    if (t == 0) s_mu = red[0] * (1.0f / DD);
    __syncthreads();
    const float mu = s_mu;

    float v = 0.0f;
#pragma unroll
    for (int u = 0; u < 3; ++u) { const float d = z[u] - mu; v += d * d; }
    red[t] = v; __syncthreads();
    for (int o = 128; o > 0; o >>= 1) { if (t < o) red[t] += red[t + o]; __syncthreads(); }
    if (t == 0) s_inv = rsqrtf(red[0] * (1.0f / DD) + 1e-12f);
    __syncthreads();
    const float inv  = s_inv;
    const float node = (mask[r] > 0.5f) ? 1.0f : 0.0f;

#pragma unroll
    for (int u = 0; u < 3; ++u) {
        const int c = t + 256 * u;
        const float o = ((z[u] - mu) * inv * g[c] + bta[c]) * node;
        out[(size_t)r * DD + c]    = o;
        hidden[(size_t)r * DD + c] = o;
    }
}

// ---------------------------------------------------------------------------
// Launch
// ---------------------------------------------------------------------------
extern "C" void kernel_launch(void* const* d_in, const int* in_sizes, int n_in,
                              void* d_out, int out_size, void* d_ws, size_t ws_size,
                              hipStream_t stream)
{
    const float* h    = (const float*)d_in[0];
    const float* msk  = (const float*)d_in[1];
    const float* Wq   = (const float*)d_in[2];
    const float* bq   = (const float*)d_in[3];
    const float* Wk   = (const float*)d_in[4];
    const float* bk   = (const float*)d_in[5];
    const float* Wv   = (const float*)d_in[6];
    const float* bv   = (const float*)d_in[7];
    const float* Wbk  = (const float*)d_in[8];   // (4,384,192)
    const float* bbk  = (const float*)d_in[9];   // (4,192)
    const float* Wbv  = (const float*)d_in[10];
    const float* bbv  = (const float*)d_in[11];
    const float* Wd1  = (const float*)d_in[12];
    const float* bd1  = (const float*)d_in[13];
    const float* WH   = (const float*)d_in[14];
    const float* bH   = (const float*)d_in[15];
    const float* WT   = (const float*)d_in[16];
    const float* bT   = (const float*)d_in[17];
    const float* Wd3  = (const float*)d_in[18];
    const float* bd3  = (const float*)d_in[19];
    const float* ln_g = (const float*)d_in[20];
    const float* ln_b = (const float*)d_in[21];

    float* w = (float*)d_ws;
    float* Qb   = w + 0 * (size_t)SEL;   // later reused as x
    float* Kbuf = w + 1 * (size_t)SEL;   // later reused as hhraw
    float* Vbuf = w + 2 * (size_t)SEL;   // later reused as traw
    float* KA   = w + 3 * (size_t)SEL;   // 4 heads x SPH ; later reused as x2
    float* KB   = w + 4 * (size_t)SEL;   //                 later reused as y
    float* VA   = w + 5 * (size_t)SEL;
    float* VB   = w + 6 * (size_t)SEL;
    float* ctx  = w + 7 * (size_t)SEL;

    const dim3 blk(256);
    const dim3 gridBig(ROWS / BM, DD / BN);    // (16, 12)
    const dim3 gridHead(ROWS / BM, HDIM / BN); // (16, 3)

    // 1) Q, K, V projections
    gemm_f32_wmma_kernel<<<gridBig, blk, 0, stream>>>(h, DD, 0, Wq, DD, bq, nullptr, Qb,   DD, ROWS, DD, DD);
    gemm_f32_wmma_kernel<<<gridBig, blk, 0, stream>>>(h, DD, 0, Wk, DD, bk, msk,     Kbuf, DD, ROWS, DD, DD);
    gemm_f32_wmma_kernel<<<gridBig, blk, 0, stream>>>(h, DD, 0, Wv, DD, bv, msk,     Vbuf, DD, ROWS, DD, DD);

    // 2) per-head pair projections (bias folded into A-part)
    for (int hd = 0; hd < HH; ++hd) {
        const float* WkA = Wbk + (size_t)hd * 2 * HDIM * HDIM;          // rows 0..191
        const float* WkB = WkA + (size_t)HDIM * HDIM;                   // rows 192..383
        const float* WvA = Wbv + (size_t)hd * 2 * HDIM * HDIM;
        const float* WvB = WvA + (size_t)HDIM * HDIM;
        gemm_f32_wmma_kernel<<<gridHead, blk, 0, stream>>>(Kbuf, DD, hd * HDIM, WkA, HDIM, bbk + hd * HDIM, nullptr, KA + (size_t)hd * SPH, HDIM, ROWS, HDIM, HDIM);
        gemm_f32_wmma_kernel<<<gridHead, blk, 0, stream>>>(Kbuf, DD, hd * HDIM, WkB, HDIM, nullptr,         nullptr, KB + (size_t)hd * SPH, HDIM, ROWS, HDIM, HDIM);
        gemm_f32_wmma_kernel<<<gridHead, blk, 0, stream>>>(Vbuf, DD, hd * HDIM, WvA, HDIM, bbv + hd * HDIM, nullptr, VA + (size_t)hd * SPH, HDIM, ROWS, HDIM, HDIM);
        gemm_f32_wmma_kernel<<<gridHead, blk, 0, stream>>>(Vbuf, DD, hd * HDIM, WvB, HDIM, nullptr,         nullptr, VB + (size_t)hd * SPH, HDIM, ROWS, HDIM, HDIM);
    }

    // 3) attention: 8192 query-heads, 8 waves (256 thr) per block
    attn_kernel<<<dim3(ROWS * HH / 8), dim3(256), 0, stream>>>(Qb, Kbuf, Vbuf, KA, KB, VA, VB, msk, ctx);

    // 4) FFN highway (reusing freed buffers)
    float* x     = Qb;
    float* hhraw = Kbuf;
    float* traw  = Vbuf;
    float* x2    = KA;
    float* y     = KB;
    gemm_f32_wmma_kernel<<<gridBig, blk, 0, stream>>>(ctx, DD, 0, Wd1, DD, bd1, nullptr, x,     DD, ROWS, DD, DD);
    gemm_f32_wmma_kernel<<<gridBig, blk, 0, stream>>>(x,   DD, 0, WH,  DD, bH,  nullptr, hhraw, DD, ROWS, DD, DD);
    gemm_f32_wmma_kernel<<<gridBig, blk, 0, stream>>>(x,   DD, 0, WT,  DD, bT,  nullptr, traw,  DD, ROWS, DD, DD);
    highway_kernel<<<dim3(SEL / 256), dim3(256), 0, stream>>>(hhraw, traw, x, x2, SEL);
    gemm_f32_wmma_kernel<<<gridBig, blk, 0, stream>>>(x2,  DD, 0, Wd3, DD, bd3, nullptr, y,     DD, ROWS, DD, DD);

    // 5) residual + LayerNorm + node mask -> (out, hidden)
    float* out    = (float*)d_out;
    float* hidden = out + (size_t)SEL;
    ln_kernel<<<dim3(ROWS), dim3(256), 0, stream>>>(y, h, msk, ln_g, ln_b, out, hidden);
}